// ACELoss_37864431682253
// MI455X (gfx1250) — compile-verified
//
#include <hip/hip_runtime.h>

// ACE targets via Cholesky whitening (mathematically identical to the SVD path):
//   C = L L^T ;  W = L^{-1} [sig | X - mean]  (128 x 33 per patch)
//   out[b,p] = <W_0, W_{b+1}> / (max(|W_0|,eps) * max(|W_{b+1}|,eps))
// One workgroup (8 wave32) per patch. Covariance streamed to LDS with
// GLOBAL_LOAD_ASYNC_TO_LDS_B128 (non-temporal: single-use 205MB stream),
// X gathered with GLOBAL_LOAD_ASYNC_TO_LDS_B32 (L2-reused across patches).
// SYRK trailing updates and the blocked triangular solve run on
// V_WMMA_F32_16X16X4_F32 (fp32 matrix cores, no precision loss).

#define PDIM 3136
#define DDIM 128
#define BDIM 32
#define LDC  132   // padded row stride for sC (132 % 64 != 0 -> no bank conflicts)
#define LDW  49    // padded row stride for sW

typedef __attribute__((ext_vector_type(2))) float v2f;
typedef __attribute__((ext_vector_type(8))) float v8f;

// byte offset of a shared-memory pointer within the workgroup LDS allocation
// (generic LDS pointers carry the LDS offset in their low 32 bits)
__device__ __forceinline__ unsigned lds_off(const void* pp) {
  return (unsigned)(uintptr_t)pp;
}

__global__ __launch_bounds__(256) void ace_whiten_kernel(
    const float* __restrict__ X, const float* __restrict__ b_mean,
    const float* __restrict__ b_cov, const float* __restrict__ sig,
    float* __restrict__ out)
{
  extern __shared__ float smem[];
  float* sC    = smem;                    // [128][132]  66.0 KB
  float* sW    = sC + DDIM * LDC;         // [128][49]   24.5 KB (cols 0..32 used, 33..47 zero)
  float* sAcc  = sW + DDIM * LDW;         // [2][16][48]  6.0 KB GEMM partial slabs
  float* sMean = sAcc + 2 * 16 * 48;      // [128]
  float* sNrm  = sMean + DDIM;            // [33]
  float* sNum  = sNrm + 33;               // [32]

  const int p    = blockIdx.x;
  const int tid  = threadIdx.x;
  const int wave = __builtin_amdgcn_readfirstlane(tid >> 5);  // SGPR -> scalar branches
  const int lane = tid & 31;
  const int row  = lane & 15;   // M (A) or N (B/D) position
  const int hi   = lane >> 4;   // half-wave selector (K pairs / M+8)

  // ---------------- Phase 0: async-stage data into LDS ----------------
  // Covariance: 64 KB contiguous -> LDS, 16B per lane per op, non-temporal.
  {
    const float* gC = b_cov + (size_t)p * (DDIM * DDIM);
    for (int i = tid; i < (DDIM * DDIM) / 4; i += 256) {       // i indexes float4s
      int r = i >> 5, c = (i & 31) * 4;                        // 32 float4 per row
      unsigned dst = lds_off(&sC[r * LDC + c]);                // 16B-aligned (LDC*4=528)
      unsigned long long src = (unsigned long long)(uintptr_t)(gC + (size_t)i * 4);
      asm volatile("global_load_async_to_lds_b128 %0, %1, off th:TH_LOAD_NT"
                   :: "v"(dst), "v"(src) : "memory");
    }
  }
  // X columns: raw values async-gathered into sW (mean subtracted in-place later).
  for (int i = tid; i < DDIM * BDIM; i += 256) {
    int d = i & 127, b = i >> 7;
    unsigned dst = lds_off(&sW[d * LDW + 1 + b]);
    unsigned long long src =
        (unsigned long long)(uintptr_t)(X + ((size_t)b * DDIM + d) * PDIM + p);
    asm volatile("global_load_async_to_lds_b32 %0, %1, off"
                 :: "v"(dst), "v"(src) : "memory");
  }
  // mean & signature columns (small strided gathers, L2-resident across blocks)
  if (tid < DDIM) {
    sMean[tid]        = b_mean[(size_t)tid * PDIM + p];
    sW[tid * LDW + 0] = sig[(size_t)tid * PDIM + p];
  }
  for (int i = tid; i < DDIM * 15; i += 256) {      // zero pad cols 33..47
    int d = i & 127, n = i >> 7;
    sW[d * LDW + 33 + n] = 0.0f;
  }
  asm volatile("s_wait_asynccnt 0" ::: "memory");   // own async ops done before barrier
  __syncthreads();
  // center X in place (needs sMean + async X data, both fenced above)
  for (int i = tid; i < DDIM * BDIM; i += 256) {
    int d = i & 127, b = i >> 7;
    sW[d * LDW + 1 + b] -= sMean[d];
  }
  __syncthreads();

  // ---------------- Phase 1: blocked Cholesky (lower) in LDS ----------------
  for (int pb = 0; pb < 8; ++pb) {
    const int j0 = pb * 16;
    // 16-column panel factorization (scalar, rank-1 restricted to panel).
    for (int jj = 0; jj < 16; ++jj) {
      const int j = j0 + jj;
      if (tid == 0) sC[j * LDC + j] = sqrtf(sC[j * LDC + j]);
      __syncthreads();
      const float dinv = 1.0f / sC[j * LDC + j];
      for (int i = j + 1 + tid; i < DDIM; i += 256) sC[i * LDC + j] *= dinv;
      __syncthreads();
      const int g = tid >> 4;        // column group within panel
      const int r = tid & 15;
      const int k = j + 1 + g;
      if (g < 15 - jj) {             // k <= j0+15
        const float ckj = sC[k * LDC + j];
        for (int i = k + r; i < DDIM; i += 16)
          sC[i * LDC + k] -= sC[i * LDC + j] * ckj;
      }
      __syncthreads();
    }
    // Trailing SYRK update: C[it,jt] -= Lp[it] * Lp[jt]^T, tiles over 8 waves.
    if (pb < 7) {
      int pi = 0;
      for (int jt = pb + 1; jt < 8; ++jt) {
        for (int it = jt; it < 8; ++it, ++pi) {
          if ((pi & 7) != wave) continue;       // scalar (wave is SGPR)
          v8f acc = {};
          #pragma unroll
          for (int c = 0; c < 4; ++c) {
            const int kb = j0 + 4 * c + 2 * hi; // A/B K placement per ISA layout
            v2f a, b;
            a.x = sC[(it * 16 + row) * LDC + kb];
            a.y = sC[(it * 16 + row) * LDC + kb + 1];
            b.x = sC[(jt * 16 + row) * LDC + kb];       // B[k][n] = Lp[jt*16+n][k]
            b.y = sC[(jt * 16 + row) * LDC + kb + 1];
            acc = __builtin_amdgcn_wmma_f32_16x16x4_f32(
                false, a, false, b, (short)0, acc, false, false);
          }
          #pragma unroll
          for (int v = 0; v < 8; ++v)
            sC[(it * 16 + v + 8 * hi) * LDC + jt * 16 + row] -= acc[v];
        }
      }
    }
    __syncthreads();
  }

  // ---------------- Phase 2: blocked solve W = L^{-1} Y ----------------
  const int nt = wave % 3;   // N-tile (waves 0..5 active in GEMM phase)
  const int h  = wave / 3;   // k-half
  for (int bi = 0; bi < 8; ++bi) {
    if (bi > 0) {
      if (wave < 6) {
        v8f acc = {};
        for (int bj = h; bj < bi; bj += 2) {
          #pragma unroll
          for (int c = 0; c < 4; ++c) {
            const int kb = 4 * c + 2 * hi;
            v2f a, b;
            a.x = sC[(bi * 16 + row) * LDC + bj * 16 + kb];
            a.y = sC[(bi * 16 + row) * LDC + bj * 16 + kb + 1];
            b.x = sW[(bj * 16 + kb    ) * LDW + nt * 16 + row];
            b.y = sW[(bj * 16 + kb + 1) * LDW + nt * 16 + row];
            acc = __builtin_amdgcn_wmma_f32_16x16x4_f32(
                false, a, false, b, (short)0, acc, false, false);
          }
        }
        #pragma unroll
        for (int v = 0; v < 8; ++v)
          sAcc[(h * 16 + v + 8 * hi) * 48 + nt * 16 + row] = acc[v];
      }
      __syncthreads();
      for (int i = tid; i < 16 * 48; i += 256) {
        int r = i / 48, n = i % 48;
        sW[(bi * 16 + r) * LDW + n] -= sAcc[r * 48 + n] + sAcc[(16 + r) * 48 + n];
      }
      __syncthreads();
    }
    // Diagonal 16x16 forward substitution, one thread per used column.
    if (tid < 33) {
      float w[16];
      #pragma unroll
      for (int r = 0; r < 16; ++r) {
        float a2 = sW[(bi * 16 + r) * LDW + tid];
        #pragma unroll
        for (int c2 = 0; c2 < 16; ++c2)
          if (c2 < r) a2 -= sC[(bi * 16 + r) * LDC + bi * 16 + c2] * w[c2];
        w[r] = a2 / sC[(bi * 16 + r) * LDC + bi * 16 + r];
        sW[(bi * 16 + r) * LDW + tid] = w[r];
      }
    }
    __syncthreads();
  }

  // ---------------- Phase 3: norms, dots, output ----------------
  if (tid < 33) {
    float s = 0.0f;
    for (int k = 0; k < DDIM; ++k) { float v = sW[k * LDW + tid]; s += v * v; }
    sNrm[tid] = s;
  }
  if (tid >= 64 && tid < 96) {
    const int b = tid - 64;
    float s = 0.0f;
    for (int k = 0; k < DDIM; ++k) s += sW[k * LDW + 0] * sW[k * LDW + 1 + b];
    sNum[b] = s;
  }
  __syncthreads();
  if (tid < BDIM) {
    const float sn = fmaxf(sqrtf(sNrm[0]), 1e-12f);
    const float xn = fmaxf(sqrtf(sNrm[tid + 1]), 1e-12f);
    out[(size_t)tid * PDIM + p] = sNum[tid] / (sn * xn);
  }
}

extern "C" void kernel_launch(void* const* d_in, const int* in_sizes, int n_in,
                              void* d_out, int out_size, void* d_ws, size_t ws_size,
                              hipStream_t stream) {
  const float* X     = (const float*)d_in[0];
  const float* bmean = (const float*)d_in[1];
  const float* bcov  = (const float*)d_in[2];
  const float* sigs  = (const float*)d_in[3];
  float* out = (float*)d_out;
  const size_t shmem =
      (size_t)(DDIM * LDC + DDIM * LDW + 2 * 16 * 48 + DDIM + 33 + 32) * sizeof(float);
  hipLaunchKernelGGL(ace_whiten_kernel, dim3(PDIM), dim3(256), shmem, stream,
                     X, bmean, bcov, sigs, out);
}